// GCNEncoder_89481348645581
// MI455X (gfx1250) — compile-verified
//
#include <hip/hip_runtime.h>

// CDNA5 / gfx1250: wave32, WMMA 16x16 tiles.
typedef __attribute__((ext_vector_type(2))) float v2f;
typedef __attribute__((ext_vector_type(8))) float v8f;

#define DIM 128

// ---------------------------------------------------------------- zero fill
__global__ __launch_bounds__(256) void k_zero(float* __restrict__ p, size_t n) {
  size_t i = (size_t)blockIdx.x * blockDim.x + threadIdx.x;
  size_t stride = (size_t)gridDim.x * blockDim.x;
  for (; i < n; i += stride) p[i] = 0.0f;
}

// ------------------------------------------------------- degree accumulation
__global__ __launch_bounds__(256) void k_deg_count(const int* __restrict__ src,
                                                   const int* __restrict__ dst,
                                                   float* __restrict__ outc,
                                                   float* __restrict__ inc,
                                                   int nE) {
  int e = blockIdx.x * blockDim.x + threadIdx.x;
  if (e < nE) {
    unsafeAtomicAdd(&outc[src[e]], 1.0f);   // -> global_atomic_add_f32
    unsafeAtomicAdd(&inc[dst[e]], 1.0f);
  }
}

__global__ __launch_bounds__(256) void k_deg_fin(float* __restrict__ outc,
                                                 float* __restrict__ inc, int n) {
  int i = blockIdx.x * blockDim.x + threadIdx.x;
  if (i < n) {
    outc[i] = rsqrtf(fmaxf(outc[i], 1.0f));
    inc[i]  = rsqrtf(fmaxf(inc[i], 1.0f));
  }
}

// --------------------------------------------------- edge-parallel scatter-add
// One wave (32 lanes) per edge; each lane moves 4 of the 128 dims as float4.
// agg[dst] += h[src] * rsqrt(out_deg[src]).  Node features (25.6 MB) are
// L2-resident on MI455X (192 MB L2), so these atomics resolve in L2.
__global__ __launch_bounds__(256) void k_scatter(const float* __restrict__ h,
                                                 const float* __restrict__ outr,
                                                 const int* __restrict__ src,
                                                 const int* __restrict__ dst,
                                                 float* __restrict__ agg,
                                                 int nE) {
  unsigned t = blockIdx.x * blockDim.x + threadIdx.x;
  unsigned e = t >> 5;
  unsigned lane = t & 31u;
  if (e >= (unsigned)nE) return;
  int s = src[e];
  int d = dst[e];
  float r = outr[s];
  const float4 v = *(const float4*)(h + (size_t)s * DIM + lane * 4);
  float* a = agg + (size_t)d * DIM + lane * 4;
  unsafeAtomicAdd(a + 0, v.x * r);
  unsafeAtomicAdd(a + 1, v.y * r);
  unsafeAtomicAdd(a + 2, v.z * r);
  unsafeAtomicAdd(a + 3, v.w * r);
}

// ----------------------------------------------------------- WMMA f32 GEMM
// out[M,128] = (agg * rsqrt(in_deg)[:,None]) @ W[128,128] + b, optional ReLU.
// One wave computes one 16x16 output tile via 32x V_WMMA_F32_16X16X4_F32.
// Block = 256 threads = 8 waves = one 16-row strip covering all 128 columns.
//
// A (16x4 f32) lane layout: lane L holds row m=L&15; VGPR v holds K = v + 2*(L>>4).
// B (4x16 f32) lane layout: lane L holds col n=L&15; VGPR v holds K = v + 2*(L>>4).
// C/D (16x16 f32): VGPR v holds row v + 8*(L>>4), col n=L&15.
__global__ __launch_bounds__(256) void k_gemm_wmma(const float* __restrict__ agg,
                                                   const float* __restrict__ inr,
                                                   const float* __restrict__ W,
                                                   const float* __restrict__ bias,
                                                   float* __restrict__ out,
                                                   int nNodes, int relu) {
  const int lane = threadIdx.x & 31;
  const int wave = threadIdx.x >> 5;        // 0..7 -> N tile index
  const int m0   = blockIdx.x * 16;         // tile row base
  const int m    = lane & 15;
  const int half = lane >> 4;               // 0 or 1
  const int gn   = wave * 16 + m;           // this lane's output column (also B col)
  int gm = m0 + m;
  const int gmc = (gm < nNodes) ? gm : (nNodes - 1);   // clamp tail rows for loads
  const float r = inr[gmc];
  const float* Arow = agg + (size_t)gmc * DIM;

  // accumulator starts at bias (broadcast down the 8 rows this lane covers)
  const float bn = bias[gn];
  v8f c;
#pragma unroll
  for (int i = 0; i < 8; i++) c[i] = bn;

#pragma unroll
  for (int k = 0; k < DIM; k += 4) {
    // A fragment: two adjacent K elements of row gmc, scaled by rsqrt(in_deg)
    v2f a = *(const v2f*)(Arow + k + 2 * half);
    a.x *= r;
    a.y *= r;
    // B fragment: W[k + 2*half + {0,1}][gn]
    const float* Wp = W + (size_t)(k + 2 * half) * DIM + gn;
    v2f bf;
    bf.x = Wp[0];
    bf.y = Wp[DIM];
    c = __builtin_amdgcn_wmma_f32_16x16x4_f32(
        /*neg_a=*/false, a, /*neg_b=*/false, bf,
        /*c_mod=*/(short)0, c, /*reuse_a=*/false, /*reuse_b=*/false);
  }

  if (relu) {
#pragma unroll
    for (int i = 0; i < 8; i++) c[i] = fmaxf(c[i], 0.0f);
  }

#pragma unroll
  for (int i = 0; i < 8; i++) {
    const int row = m0 + i + 8 * half;
    if (row < nNodes) out[(size_t)row * DIM + gn] = c[i];
  }
}

// ------------------------------------------------------------------ launcher
extern "C" void kernel_launch(void* const* d_in, const int* in_sizes, int n_in,
                              void* d_out, int out_size, void* d_ws, size_t ws_size,
                              hipStream_t stream) {
  const float* feat = (const float*)d_in[0];
  const int*   src  = (const int*)d_in[1];
  const int*   dst  = (const int*)d_in[2];
  const float* Ws[3] = {(const float*)d_in[3], (const float*)d_in[5], (const float*)d_in[7]};
  const float* bs[3] = {(const float*)d_in[4], (const float*)d_in[6], (const float*)d_in[8]};

  const int nNodes = in_sizes[0] / DIM;
  const int nEdges = in_sizes[1];
  const size_t padN = ((size_t)nNodes + 63) & ~(size_t)63;

  // d_ws layout: [outr | inr | cur | agg]
  float* outr = (float*)d_ws;
  float* inr  = outr + padN;
  float* cur  = inr + padN;
  float* agg  = cur + (size_t)nNodes * DIM;

  const int nFeat = nNodes * DIM;

  // degrees (same for all 3 layers)
  k_zero<<<512, 256, 0, stream>>>(outr, 2 * padN);
  k_deg_count<<<(nEdges + 255) / 256, 256, 0, stream>>>(src, dst, outr, inr, nEdges);
  k_deg_fin<<<(nNodes + 255) / 256, 256, 0, stream>>>(outr, inr, nNodes);

  const float* in_feat = feat;
  const unsigned scatterThreads = (unsigned)nEdges * 32u;
  for (int l = 0; l < 3; l++) {
    k_zero<<<2048, 256, 0, stream>>>(agg, (size_t)nFeat);
    k_scatter<<<(scatterThreads + 255) / 256, 256, 0, stream>>>(in_feat, outr, src, dst, agg, nEdges);
    float* o = (l == 2) ? (float*)d_out : cur;
    k_gemm_wmma<<<(nNodes + 15) / 16, 256, 0, stream>>>(agg, inr, Ws[l], bs[l], o, nNodes, (l < 2) ? 1 : 0);
    in_feat = cur;   // GEMM output overwrote cur (its input was already consumed)
  }
}